// GraphEncoder_7395933683920
// MI455X (gfx1250) — compile-verified
//
#include <hip/hip_runtime.h>
#include <hip/hip_bf16.h>

#define D_DIM 128
#define H_HEADS 8
#define O_FEATS 16
#define SCAN_T 1024

typedef __attribute__((ext_vector_type(2))) float v2f;
typedef __attribute__((ext_vector_type(8))) float v8f;

// ---------------------------------------------------------------------------
__global__ void fill_i32(int* __restrict__ p, int v, int n) {
    int i = blockIdx.x * blockDim.x + threadIdx.x;
    if (i < n) p[i] = v;
}

// ---------------------------------------------------------------------------
// Row gather: out[i][:] = src[idx[i]][:]   (128 floats per row, float4 lanes)
__global__ void gather_rows(const float* __restrict__ src,
                            const int* __restrict__ idx,
                            float* __restrict__ out, int nrows) {
    int t = blockIdx.x * blockDim.x + threadIdx.x;
    if (t >= nrows * 32) return;
    int i = t >> 5;
    int c = (t & 31) * 4;
    const float4 v = *(const float4*)(src + (size_t)idx[i] * D_DIM + c);
    *(float4*)(out + (size_t)i * D_DIM + c) = v;
}

// ---------------------------------------------------------------------------
// Z = A @ W  with A: nrows x 128, W: 128 x 128, via V_WMMA_F32_16X16X4_F32.
// blockDim = (32, 8): one wave per 16x16 output tile; threadIdx.y = N-tile.
__global__ void __launch_bounds__(256)
gemm_wmma_f32(const float* __restrict__ A, const float* __restrict__ W,
              float* __restrict__ Z, int nrows) {
    const int lane  = threadIdx.x;        // 0..31
    const int ntile = threadIdx.y;        // 0..7
    const int mbase = blockIdx.x * 16;
    const int half  = lane >> 4;          // 0: K+0/1, 1: K+2/3
    const int l16   = lane & 15;
    const int row   = min(mbase + l16, nrows - 1);   // clamp; masked at store
    const int col   = ntile * 16 + l16;

    const float* __restrict__ arow = A + (size_t)row * D_DIM;

    v8f c = {};
    #pragma unroll
    for (int kt = 0; kt < 32; ++kt) {
        const int k = kt * 4 + half * 2;
        v2f a;
        a.x = arow[k];
        a.y = arow[k + 1];
        v2f b;
        b.x = W[(size_t)k * D_DIM + col];
        b.y = W[(size_t)(k + 1) * D_DIM + col];
        // D = A(16x4) * B(4x16) + C ; fp32 in, fp32 accumulate
        c = __builtin_amdgcn_wmma_f32_16x16x4_f32(false, a, false, b,
                                                  (short)0, c, false, false);
    }

    #pragma unroll
    for (int v = 0; v < 8; ++v) {
        const int m = mbase + v + half * 8;    // C/D layout: M = vgpr + 8*(lane/16)
        if (m < nrows) Z[(size_t)m * D_DIM + col] = c[v];
    }
}

// ---------------------------------------------------------------------------
// el[n,h] = dot(z[n,h,:], al[h,:]) ; er likewise. One thread per (n,h).
__global__ void attn_proj(const float* __restrict__ z,
                          const float* __restrict__ al,
                          const float* __restrict__ ar,
                          float* __restrict__ el, float* __restrict__ er,
                          int nrows) {
    int idx = blockIdx.x * blockDim.x + threadIdx.x;
    if (idx >= nrows * H_HEADS) return;
    int n = idx >> 3, h = idx & 7;
    const float* zp  = z  + (size_t)n * D_DIM + h * O_FEATS;
    const float* alp = al + h * O_FEATS;
    const float* arp = ar + h * O_FEATS;
    float sl = 0.0f, sr = 0.0f;
    #pragma unroll
    for (int o = 0; o < O_FEATS; ++o) {
        float zv = zp[o];
        sl += zv * alp[o];
        sr += zv * arp[o];
    }
    el[idx] = sl;
    er[idx] = sr;
}

// ---------------------------------------------------------------------------
// CSR build: histogram of in-degrees
__global__ void hist_dst(const int* __restrict__ dst, int* __restrict__ deg, int nE) {
    int e = blockIdx.x * blockDim.x + threadIdx.x;
    if (e < nE) atomicAdd(&deg[dst[e]], 1);
}

// Single-workgroup chunked exclusive scan; writes out[0..n-1] and total at out[n].
__global__ void __launch_bounds__(SCAN_T)
scan_excl(const int* __restrict__ in, int* __restrict__ out, int n) {
    __shared__ int buf[SCAN_T];
    __shared__ int carry;
    if (threadIdx.x == 0) carry = 0;
    __syncthreads();
    for (int base = 0; base < n; base += SCAN_T) {
        int i = base + (int)threadIdx.x;
        int v = (i < n) ? in[i] : 0;
        buf[threadIdx.x] = v;
        __syncthreads();
        #pragma unroll
        for (int off = 1; off < SCAN_T; off <<= 1) {
            int t = (threadIdx.x >= off) ? buf[threadIdx.x - off] : 0;
            __syncthreads();
            buf[threadIdx.x] += t;
            __syncthreads();
        }
        int excl = buf[threadIdx.x] - v + carry;
        if (i < n) out[i] = excl;
        __syncthreads();
        if (threadIdx.x == SCAN_T - 1) carry += buf[SCAN_T - 1];
        __syncthreads();
    }
    if (threadIdx.x == 0) out[n] = carry;
}

__global__ void copy_i32(const int* __restrict__ a, int* __restrict__ b, int n) {
    int i = blockIdx.x * blockDim.x + threadIdx.x;
    if (i < n) b[i] = a[i];
}

// Bucket edge ids by destination node (order within bucket is arbitrary;
// max/sum/weighted-sum are order-insensitive up to fp rounding).
__global__ void scatter_eids(const int* __restrict__ dst, int* __restrict__ cursor,
                             int* __restrict__ eids, int nE) {
    int e = blockIdx.x * blockDim.x + threadIdx.x;
    if (e >= nE) return;
    int p = atomicAdd(&cursor[dst[e]], 1);
    eids[p] = e;
}

// ---------------------------------------------------------------------------
// Atomic-free GAT aggregation: one wave per destination node.
//   scores -> per-head max -> exp-sum -> alpha-weighted sum of z[src]
//   fused epilogue: + h_in + bias (+ leaky_relu 0.01 on layer 0)
__global__ void __launch_bounds__(256)
gat_aggregate(const int* __restrict__ rowstart, const int* __restrict__ eids,
              const int* __restrict__ srcv,
              const float* __restrict__ el, const float* __restrict__ er,
              const float* __restrict__ z, const float* __restrict__ hin,
              const float* __restrict__ bias, float* __restrict__ hout,
              int nrows, int act) {
    const int node = blockIdx.x * (blockDim.x >> 5) + ((int)threadIdx.x >> 5);
    const int lane = threadIdx.x & 31;
    if (node >= nrows) return;                     // wave-uniform exit

    const int beg = rowstart[node];
    const int end = rowstart[node + 1];

    // er[node][0..7] (two float4 loads; broadcast across wave)
    const float4 er0 = *(const float4*)(er + (size_t)node * H_HEADS);
    const float4 er1 = *(const float4*)(er + (size_t)node * H_HEADS + 4);
    float erd[8] = {er0.x, er0.y, er0.z, er0.w, er1.x, er1.y, er1.z, er1.w};

    // ---- Phase 1: per-head max over incoming edges (lanes stride edges) ----
    float mh[8];
    #pragma unroll
    for (int h = 0; h < 8; ++h) mh[h] = -INFINITY;
    for (int j = beg + lane; j < end; j += 32) {
        const int s = srcv[eids[j]];
        const float4 a0 = *(const float4*)(el + (size_t)s * H_HEADS);
        const float4 a1 = *(const float4*)(el + (size_t)s * H_HEADS + 4);
        const float ev[8] = {a0.x, a0.y, a0.z, a0.w, a1.x, a1.y, a1.z, a1.w};
        #pragma unroll
        for (int h = 0; h < 8; ++h) {
            float x = ev[h] + erd[h];
            x = (x >= 0.0f) ? x : 0.2f * x;
            mh[h] = fmaxf(mh[h], x);
        }
    }
    #pragma unroll
    for (int h = 0; h < 8; ++h)
        #pragma unroll
        for (int off = 16; off > 0; off >>= 1)
            mh[h] = fmaxf(mh[h], __shfl_xor(mh[h], off, 32));

    // ---- Phase 2: per-head exp-sum ----
    float sh[8];
    #pragma unroll
    for (int h = 0; h < 8; ++h) sh[h] = 0.0f;
    for (int j = beg + lane; j < end; j += 32) {
        const int s = srcv[eids[j]];
        const float4 a0 = *(const float4*)(el + (size_t)s * H_HEADS);
        const float4 a1 = *(const float4*)(el + (size_t)s * H_HEADS + 4);
        const float ev[8] = {a0.x, a0.y, a0.z, a0.w, a1.x, a1.y, a1.z, a1.w};
        #pragma unroll
        for (int h = 0; h < 8; ++h) {
            float x = ev[h] + erd[h];
            x = (x >= 0.0f) ? x : 0.2f * x;
            sh[h] += __expf(x - mh[h]);
        }
    }
    #pragma unroll
    for (int h = 0; h < 8; ++h)
        #pragma unroll
        for (int off = 16; off > 0; off >>= 1)
            sh[h] += __shfl_xor(sh[h], off, 32);

    // ---- Phase 3: alpha-weighted gather of z[src]; lane -> (head, 4 feats) ----
    const int head = lane >> 2;
    const int o    = (lane & 3) * 4;
    const float m_h  = mh[head];
    const float s_h  = sh[head];
    const float er_h = erd[head];
    float4 acc = {0.0f, 0.0f, 0.0f, 0.0f};
    for (int j = beg; j < end; ++j) {            // wave-uniform loop (~deg iters)
        const int s = srcv[eids[j]];
        float x = el[(size_t)s * H_HEADS + head] + er_h;
        x = (x >= 0.0f) ? x : 0.2f * x;
        const float alpha = __expf(x - m_h) / s_h;
        const float4 zv = *(const float4*)(z + (size_t)s * D_DIM + head * O_FEATS + o);
        acc.x += alpha * zv.x;
        acc.y += alpha * zv.y;
        acc.z += alpha * zv.z;
        acc.w += alpha * zv.w;
    }

    // ---- Epilogue: residual + bias (+ activation) ----
    const size_t base = (size_t)node * D_DIM + head * O_FEATS + o;
    const float4 hv = *(const float4*)(hin + base);
    const float4 bv = *(const float4*)(bias + head * O_FEATS + o);
    float4 r;
    r.x = acc.x + hv.x + bv.x;
    r.y = acc.y + hv.y + bv.y;
    r.z = acc.z + hv.z + bv.z;
    r.w = acc.w + hv.w + bv.w;
    if (act) {
        r.x = (r.x >= 0.0f) ? r.x : 0.01f * r.x;
        r.y = (r.y >= 0.0f) ? r.y : 0.01f * r.y;
        r.z = (r.z >= 0.0f) ? r.z : 0.01f * r.z;
        r.w = (r.w >= 0.0f) ? r.w : 0.01f * r.w;
    }
    *(float4*)(hout + base) = r;
}

// ---------------------------------------------------------------------------
// In-place L2 normalize each 128-float row: v = h / max(||h||, 1e-5)
__global__ void l2norm(float* __restrict__ h, int nrows) {
    int n = blockIdx.x * blockDim.x + threadIdx.x;
    if (n >= nrows) return;
    float* p = h + (size_t)n * D_DIM;
    float ss = 0.0f;
    #pragma unroll 4
    for (int i = 0; i < D_DIM; ++i) ss += p[i] * p[i];
    float inv = 1.0f / fmaxf(sqrtf(ss), 1e-5f);
    #pragma unroll 4
    for (int i = 0; i < D_DIM; ++i) p[i] *= inv;
}

// ---------------------------------------------------------------------------
static inline int ceil_div(int a, int b) { return (a + b - 1) / b; }

extern "C" void kernel_launch(void* const* d_in, const int* in_sizes, int n_in,
                              void* d_out, int out_size, void* d_ws, size_t ws_size,
                              hipStream_t stream) {
    // Inputs (setup_inputs order):
    const float* emb = (const float*)d_in[0];
    const float* W0  = (const float*)d_in[1];
    const float* al0 = (const float*)d_in[2];
    const float* ar0 = (const float*)d_in[3];
    const float* b0  = (const float*)d_in[4];
    const float* W1  = (const float*)d_in[5];
    const float* al1 = (const float*)d_in[6];
    const float* ar1 = (const float*)d_in[7];
    const float* b1  = (const float*)d_in[8];
    const int* n_feat = (const int*)d_in[9];
    const int* src    = (const int*)d_in[10];
    const int* dst    = (const int*)d_in[11];
    const int* x      = (const int*)d_in[12];

    const int N  = in_sizes[9];          // 57254
    const int E  = in_sizes[10];         // 916064
    const int BT = in_sizes[12];         // 64*512

    // Workspace carve-out: floats first, then ints
    float* ws = (float*)d_ws;
    float* hA = ws;                          // N x 128 (layer input / final v)
    float* z  = hA + (size_t)N * D_DIM;      // N x 128
    float* hB = z  + (size_t)N * D_DIM;      // N x 128 (layer output)
    float* el = hB + (size_t)N * D_DIM;      // N x 8
    float* er = el + (size_t)N * H_HEADS;    // N x 8
    int* ib       = (int*)(er + (size_t)N * H_HEADS);
    int* deg      = ib;                      // N   (degree hist, then cursor)
    int* rowstart = deg + N;                 // N+1 (CSR row offsets)
    int* eids     = rowstart + (N + 1);      // E   (edge ids bucketed by dst)

    const int TB = 256;
    const int NH = N * H_HEADS;

    // h = emb[n_feat]
    gather_rows<<<ceil_div(N * 32, TB), TB, 0, stream>>>(emb, n_feat, hA, N);

    // ---- Build CSR-by-destination (once per call; reused by both layers) ----
    fill_i32<<<ceil_div(N, TB), TB, 0, stream>>>(deg, 0, N);
    hist_dst<<<ceil_div(E, TB), TB, 0, stream>>>(dst, deg, E);
    scan_excl<<<1, SCAN_T, 0, stream>>>(deg, rowstart, N);
    copy_i32<<<ceil_div(N, TB), TB, 0, stream>>>(rowstart, deg, N);   // deg -> cursor
    scatter_eids<<<ceil_div(E, TB), TB, 0, stream>>>(dst, deg, eids, E);

    const float* Ws[2]  = {W0, W1};
    const float* als[2] = {al0, al1};
    const float* ars[2] = {ar0, ar1};
    const float* bs[2]  = {b0, b1};
    float* hin  = hA;
    float* hout = hB;

    for (int layer = 0; layer < 2; ++layer) {
        // z = h @ W   (WMMA f32 16x16x4, K=128)
        gemm_wmma_f32<<<dim3(ceil_div(N, 16)), dim3(32, 8), 0, stream>>>(hin, Ws[layer], z, N);

        // el/er projections
        attn_proj<<<ceil_div(NH, TB), TB, 0, stream>>>(z, als[layer], ars[layer], el, er, N);

        // atomic-free segment softmax + aggregation + residual + bias (+act)
        gat_aggregate<<<ceil_div(N, TB / 32), TB, 0, stream>>>(
            rowstart, eids, src, el, er, z, hin, bs[layer], hout, N,
            layer == 0 ? 1 : 0);

        float* tmp = hin; hin = hout; hout = tmp;
    }
    // after 2 swaps the final layer output is in hA (== hin now)
    l2norm<<<ceil_div(N, TB), TB, 0, stream>>>(hin, N);

    // out[b,t,:] = v[x[b,t],:]
    gather_rows<<<ceil_div(BT * 32, TB), TB, 0, stream>>>(hin, x, (float*)d_out, BT);
}